// GATLayer_807453852008
// MI455X (gfx1250) — compile-verified
//
#include <hip/hip_runtime.h>
#include <hip/hip_bf16.h>

#define N_NODES 10000
#define N_EDGES 320000
#define IN_DIM  512
#define OUT_DIM 256
#define NEG_SLOPE 0.01f

typedef float v2f __attribute__((ext_vector_type(2)));
typedef float v8f __attribute__((ext_vector_type(8)));

// ---- order-preserving float<->uint encoding for atomicMax-based segment max ----
__device__ __forceinline__ unsigned enc_f32(float f) {
    int b = __float_as_int(f);
    return (b >= 0) ? ((unsigned)b | 0x80000000u) : ~(unsigned)b;
}
__device__ __forceinline__ float dec_f32(unsigned k) {
    int b = (k & 0x80000000u) ? (int)(k & 0x7FFFFFFFu) : ~(int)k;
    return __int_as_float(b);
}
#define EMAX_NEG_INF 0x007FFFFFu   // enc_f32(-inf)

// ---------------- init: emax = enc(-inf), denom = 0, h_out = 0 ----------------
__global__ void k_init(unsigned* __restrict__ emax, float* __restrict__ denom,
                       float* __restrict__ hout) {
    int i = blockIdx.x * blockDim.x + threadIdx.x;
    if (i < N_NODES) { emax[i] = EMAX_NEG_INF; denom[i] = 0.0f; }
    for (int j = i; j < N_NODES * OUT_DIM; j += gridDim.x * blockDim.x)
        hout[j] = 0.0f;
}

// ---------------- GEMM: z[n,o] = sum_k h[n,k] * w[o,k] + b[o] -----------------
// One wave computes a 16(M) x 64(N) strip with 4 f32 accumulators:
// the A fragment is loaded once per k-step and fed to 4 WMMAs (5 loads / 4 wmma).
// V_WMMA_F32_16X16X4_F32 fragment layouts:
//   A (16x4): lane m=lane&15, kb=(lane>>4)*2 -> {A[m][k+kb], A[m][k+kb+1]}
//   B (4x16): lane n=lane&15, same kb        -> {w[o0+n][k+kb], w[o0+n][k+kb+1]}
//   D (16x16): VGPR v, lanes 0-15 -> M=v, lanes 16-31 -> M=v+8, N=lane&15
__global__ void k_gemm_wmma(const float* __restrict__ h, const float* __restrict__ w,
                            const float* __restrict__ bias, float* __restrict__ z) {
    const int lane = threadIdx.x & 31;
    const int gw   = blockIdx.x * (blockDim.x >> 5) + (threadIdx.x >> 5);
    if (gw >= 2500) return;            // 625 M-tiles * 4 N-supertiles
    const int row0 = (gw >> 2) * 16;   // 10000 = 625*16, exact
    const int col0 = (gw & 3) * 64;    // 256 = 4*64, exact
    const int mn   = lane & 15;
    const int kb   = (lane >> 4) << 1; // 0 or 2

    const float* arow = h + (size_t)(row0 + mn) * IN_DIM + kb;
    const float* b0   = w + (size_t)(col0 + mn) * IN_DIM + kb;
    const float* b1   = b0 + (size_t)16 * IN_DIM;
    const float* b2   = b0 + (size_t)32 * IN_DIM;
    const float* b3   = b0 + (size_t)48 * IN_DIM;

    v8f c0 = {}, c1 = {}, c2 = {}, c3 = {};
#pragma unroll 4
    for (int k = 0; k < IN_DIM; k += 4) {
        v2f a  = *(const v2f*)(arow + k);
        v2f q0 = *(const v2f*)(b0 + k);
        v2f q1 = *(const v2f*)(b1 + k);
        v2f q2 = *(const v2f*)(b2 + k);
        v2f q3 = *(const v2f*)(b3 + k);
        c0 = __builtin_amdgcn_wmma_f32_16x16x4_f32(false, a, false, q0, (short)0, c0, false, false);
        c1 = __builtin_amdgcn_wmma_f32_16x16x4_f32(false, a, false, q1, (short)0, c1, false, false);
        c2 = __builtin_amdgcn_wmma_f32_16x16x4_f32(false, a, false, q2, (short)0, c2, false, false);
        c3 = __builtin_amdgcn_wmma_f32_16x16x4_f32(false, a, false, q3, (short)0, c3, false, false);
    }

    const int mbase = (lane >> 4) * 8;
    v8f acc[4] = {c0, c1, c2, c3};
#pragma unroll
    for (int g = 0; g < 4; ++g) {
        const int   col = col0 + g * 16 + mn;
        const float bv  = bias[col];
#pragma unroll
        for (int v = 0; v < 8; ++v)
            z[(size_t)(row0 + mbase + v) * OUT_DIM + col] = acc[g][v] + bv;
    }
}

// ------------- per-node scalars: s_src[n]=z[n]·a_src, s_dst[n]=z[n]·a_dst -----
__global__ void k_node_dots(const float* __restrict__ z, const float* __restrict__ attn_w,
                            float* __restrict__ s_src, float* __restrict__ s_dst) {
    const int lane = threadIdx.x & 31;
    const int node = (blockIdx.x * blockDim.x + threadIdx.x) >> 5;
    if (node >= N_NODES) return;
    const float4* zrow = (const float4*)(z + (size_t)node * OUT_DIM);
    const float4* aw   = (const float4*)attn_w;           // [0..63] = a_src, [64..127] = a_dst
    float ps = 0.0f, pd = 0.0f;
#pragma unroll
    for (int j = 0; j < 2; ++j) {
        int vi = lane + j * 32;                           // float4 index within row
        float4 zz = zrow[vi];
        float4 as = aw[vi];
        float4 ad = aw[64 + vi];
        ps += zz.x * as.x + zz.y * as.y + zz.z * as.z + zz.w * as.w;
        pd += zz.x * ad.x + zz.y * ad.y + zz.z * ad.z + zz.w * ad.w;
    }
#pragma unroll
    for (int m = 16; m >= 1; m >>= 1) {
        ps += __shfl_xor(ps, m, 32);
        pd += __shfl_xor(pd, m, 32);
    }
    if (lane == 0) { s_src[node] = ps; s_dst[node] = pd; }
}

// ------------- edge logits + leaky-relu + segment max via atomicMax ----------
__global__ void k_edge_scores(const int* __restrict__ src, const int* __restrict__ dst,
                              const float* __restrict__ s_src, const float* __restrict__ s_dst,
                              const float* __restrict__ attn_b,
                              float* __restrict__ e, unsigned* __restrict__ emax) {
    int i = blockIdx.x * blockDim.x + threadIdx.x;
    if (i >= N_EDGES) return;
    float v = s_src[src[i]] + s_dst[dst[i]] + attn_b[0];
    v = (v >= 0.0f) ? v : NEG_SLOPE * v;
    e[i] = v;
    atomicMax(&emax[dst[i]], enc_f32(v));
}

// ------------- e_exp = exp(e - emax[dst]); denom[dst] += e_exp ----------------
__global__ void k_edge_exp(const int* __restrict__ dst, float* __restrict__ e,
                           const unsigned* __restrict__ emax, float* __restrict__ denom) {
    int i = blockIdx.x * blockDim.x + threadIdx.x;
    if (i >= N_EDGES) return;
    int d = dst[i];
    float ee = expf(e[i] - dec_f32(emax[d]));
    e[i] = ee;
    unsafeAtomicAdd(&denom[d], ee);
}

// ------------- h_out[dst] += (e_exp/denom[dst]) * z[src]  (wave per edge) -----
__global__ void k_aggregate(const int* __restrict__ src, const int* __restrict__ dst,
                            const float* __restrict__ e_exp, const float* __restrict__ denom,
                            const float* __restrict__ z, float* __restrict__ hout) {
    const int lane = threadIdx.x & 31;
    const int edge = (blockIdx.x * blockDim.x + threadIdx.x) >> 5;
    if (edge >= N_EDGES) return;
    const int s = src[edge], d = dst[edge];
    const float alpha = e_exp[edge] / denom[d];
    const float4* zs = (const float4*)(z + (size_t)s * OUT_DIM);
    float* ho = hout + (size_t)d * OUT_DIM;
#pragma unroll
    for (int j = 0; j < 2; ++j) {
        int vi = lane + j * 32;            // float4 index; b128 load, 4 scalar atomics
        float4 zz = zs[vi];
        unsafeAtomicAdd(&ho[vi * 4 + 0], alpha * zz.x);
        unsafeAtomicAdd(&ho[vi * 4 + 1], alpha * zz.y);
        unsafeAtomicAdd(&ho[vi * 4 + 2], alpha * zz.z);
        unsafeAtomicAdd(&ho[vi * 4 + 3], alpha * zz.w);
    }
}

static inline size_t align256(size_t x) { return (x + 255) & ~(size_t)255; }

extern "C" void kernel_launch(void* const* d_in, const int* in_sizes, int n_in,
                              void* d_out, int out_size, void* d_ws, size_t ws_size,
                              hipStream_t stream) {
    const float* h      = (const float*)d_in[0];
    const float* fc_w   = (const float*)d_in[1];
    const float* fc_b   = (const float*)d_in[2];
    const float* attn_w = (const float*)d_in[3];
    const float* attn_b = (const float*)d_in[4];
    const int*   src    = (const int*)d_in[5];
    const int*   dst    = (const int*)d_in[6];
    float* hout = (float*)d_out;

    char* ws = (char*)d_ws;
    float*    z      = (float*)ws;    ws += align256((size_t)N_NODES * OUT_DIM * 4);
    float*    s_src  = (float*)ws;    ws += align256((size_t)N_NODES * 4);
    float*    s_dst  = (float*)ws;    ws += align256((size_t)N_NODES * 4);
    float*    e_buf  = (float*)ws;    ws += align256((size_t)N_EDGES * 4);
    unsigned* emax   = (unsigned*)ws; ws += align256((size_t)N_NODES * 4);
    float*    denom  = (float*)ws;    ws += align256((size_t)N_NODES * 4);

    // 1) init accumulators
    k_init<<<1250, 256, 0, stream>>>(emax, denom, hout);
    // 2) z = h @ Wt + b   (2500 waves, 16x64 strip each, 8 waves/block)
    k_gemm_wmma<<<313, 256, 0, stream>>>(h, fc_w, fc_b, z);
    // 3) per-node attention scalars (wave per node)
    k_node_dots<<<1250, 256, 0, stream>>>(z, attn_w, s_src, s_dst);
    // 4) edge logits + segment max
    k_edge_scores<<<(N_EDGES + 255) / 256, 256, 0, stream>>>(src, dst, s_src, s_dst,
                                                             attn_b, e_buf, emax);
    // 5) exp + denominator
    k_edge_exp<<<(N_EDGES + 255) / 256, 256, 0, stream>>>(dst, e_buf, emax, denom);
    // 6) weighted scatter-add (wave per edge)
    k_aggregate<<<(N_EDGES * 32 + 255) / 256, 256, 0, stream>>>(src, dst, e_buf, denom,
                                                                z, hout);
}